// ChamferLoss_40226663694451
// MI455X (gfx1250) — compile-verified
//
#include <hip/hip_runtime.h>

// Chamfer loss for [4,8192,3] point clouds on gfx1250.
// sq-dist tile = one V_WMMA_F32_16X16X4_F32:
//   A row (16x4): [x0, x1, x2, 1]
//   B col (4x16): [-2y0, -2y1, -2y2, |y|^2]   (precomputed in WMMA-B lane layout)
//   C (16x16):    broadcast |x|^2 per row (loop invariant per row-block)
// D = |x|^2 + |y|^2 - 2 x.y  -> running min in VGPRs, sqrt deferred to finalize.

typedef float v2f __attribute__((ext_vector_type(2)));
typedef float v4f __attribute__((ext_vector_type(4)));
typedef float v8f __attribute__((ext_vector_type(8)));

#define TILE 16
#define CHUNKS 4  // column chunks per row-block -> 8192 waves per pass

__global__ __launch_bounds__(256) void chamfer_init(unsigned int* minbuf, int total,
                                                    float* out) {
  int i = blockIdx.x * blockDim.x + threadIdx.x;
  if (i < total) minbuf[i] = 0x7F7FFFFFu;  // +FLT_MAX bit pattern
  if (i == 0) out[0] = 0.0f;
}

// Per point j, emit the two B-operand VGPR pairs in lane layout:
//   [0..1] = (-2y0, -2y1)  (lanes 0-15: K0,K1)   [2..3] = (-2y2, |y|^2) (lanes 16-31: K2,K3)
__global__ __launch_bounds__(256) void chamfer_prep(const float* __restrict__ src,
                                                    float* __restrict__ dst, int npts) {
  int i = blockIdx.x * blockDim.x + threadIdx.x;
  if (i >= npts) return;
  float c0 = src[(size_t)i * 3 + 0];
  float c1 = src[(size_t)i * 3 + 1];
  float c2 = src[(size_t)i * 3 + 2];
  float n = fmaf(c2, c2, fmaf(c1, c1, c0 * c0));
  v4f o;
  o.x = -2.0f * c0;
  o.y = -2.0f * c1;
  o.z = -2.0f * c2;
  o.w = n;
  ((v4f*)dst)[i] = o;  // global_store_b128
}

// One wave handles (batch b, 16-row block rb, column chunk).
// X: raw rows [B,Nx,3]; Ytr: precomputed B-operands [B*Ny,4].
__global__ __launch_bounds__(256) void chamfer_min_pass(
    const float* __restrict__ X, const float* __restrict__ Ytr,
    unsigned int* __restrict__ minbuf, int Nx, int Ny, int rowBlocksPerBatch) {
  const int lane = threadIdx.x & 31;
  const int gwave = blockIdx.x * (blockDim.x >> 5) + (threadIdx.x >> 5);
  const int chunk = gwave % CHUNKS;
  const int rb = (gwave / CHUNKS) % rowBlocksPerBatch;
  const int b = gwave / (CHUNKS * rowBlocksPerBatch);

  const int lane15 = lane & 15;
  const bool lo = lane < 16;
  const int hsel = (lane >> 4) << 3;  // 0 for lanes 0-15, 8 for lanes 16-31

  // ---- A matrix + C (row-norm broadcast) setup: once per wave ----
  const int rowBase = rb * TILE;
  const float* xp = X + ((size_t)b * Nx + rowBase + lane15) * 3;
  const float x0 = xp[0], x1 = xp[1], x2 = xp[2];
  const float xn = fmaf(x2, x2, fmaf(x1, x1, x0 * x0));
  v2f A;
  A.x = lo ? x0 : x2;    // half0: K=0 ; half1: K=2
  A.y = lo ? x1 : 1.0f;  // half0: K=1 ; half1: K=3 (pad = 1 -> picks up |y|^2 row of B)

  // C/D layout: VGPR v, lane l -> row = v + 8*(l>=16), col = l&15.
  v8f C;
#pragma unroll
  for (int v = 0; v < 8; ++v) ((float*)&C)[v] = __shfl(xn, v + hsel, 32);

  // ---- inner loop over 16-column tiles of this chunk ----
  const int colsPerChunk = Ny / CHUNKS;
  const int tiles = colsPerChunk / TILE;
  // Each lane loads exactly its B-operand pair: one global_load_b64 per tile.
  const v2f* yb = (const v2f*)Ytr +
                  ((size_t)b * Ny + (size_t)chunk * colsPerChunk + lane15) * 2 +
                  (lo ? 0 : 1);

  v8f rmin;
#pragma unroll
  for (int v = 0; v < 8; ++v) ((float*)&rmin)[v] = 3.0e38f;

  for (int t = 0; t < tiles; ++t) {
    v2f Bm = yb[(size_t)t * TILE * 2];  // advance 16 points * 2 pairs per tile

    v8f D = __builtin_amdgcn_wmma_f32_16x16x4_f32(
        /*neg_a=*/false, A, /*neg_b=*/false, Bm,
        /*c_mod=*/(short)0, C, /*reuse_a=*/false, /*reuse_b=*/false);

#pragma unroll
    for (int v = 0; v < 8; ++v)
      ((float*)&rmin)[v] = fminf(((float*)&rmin)[v], ((float*)&D)[v]);
  }

  // ---- per-row min: reduce across the 16 lanes of each half ----
#pragma unroll
  for (int v = 0; v < 8; ++v) {
    float m = ((float*)&rmin)[v];
    m = fminf(m, __shfl_xor(m, 1, 32));
    m = fminf(m, __shfl_xor(m, 2, 32));
    m = fminf(m, __shfl_xor(m, 4, 32));
    m = fminf(m, __shfl_xor(m, 8, 32));
    ((float*)&rmin)[v] = m;
  }

  // lane 0 owns rows 0..7, lane 16 owns rows 8..15 -> one atomic instr serves both.
  // Values clamped >= 0, so uint ordering == float ordering.
  if (lane15 == 0) {
    unsigned int* mb = minbuf + (size_t)b * Nx + rowBase + hsel;
#pragma unroll
    for (int v = 0; v < 8; ++v) {
      float m = fmaxf(((float*)&rmin)[v], 0.0f);
      atomicMin(mb + v, __float_as_uint(m));
    }
  }
}

__global__ __launch_bounds__(256) void chamfer_finalize(
    const unsigned int* __restrict__ minbuf, float* __restrict__ out,
    int n1, int total, float s1, float s2) {
  int i = blockIdx.x * blockDim.x + threadIdx.x;
  float val = 0.0f;
  if (i < total) {
    float f = __uint_as_float(minbuf[i]);
    val = sqrtf(fmaxf(f, 0.0f)) * ((i < n1) ? s1 : s2);
  }
#pragma unroll
  for (int off = 16; off > 0; off >>= 1) val += __shfl_xor(val, off, 32);
  if ((threadIdx.x & 31) == 0) atomicAdd(out, val);
}

extern "C" void kernel_launch(void* const* d_in, const int* in_sizes, int n_in,
                              void* d_out, int out_size, void* d_ws, size_t ws_size,
                              hipStream_t stream) {
  const float* pred = (const float*)d_in[0];    // [B,N,3] f32
  const float* target = (const float*)d_in[1];  // [B,M,3] f32
  float* out = (float*)d_out;                   // scalar f32

  const int B = 4;
  const int N = in_sizes[0] / (B * 3);
  const int M = in_sizes[1] / (B * 3);

  // d_ws layout: [B*(N+M)] uint minbuf | [B*N*4] f32 pred-trans | [B*M*4] f32 target-trans
  unsigned int* minbuf = (unsigned int*)d_ws;
  const int total = B * (N + M);
  float* ptr_trans = (float*)((char*)d_ws + (size_t)total * sizeof(unsigned int));
  float* tgt_trans = ptr_trans + (size_t)B * N * 4;

  chamfer_init<<<(total + 255) / 256, 256, 0, stream>>>(minbuf, total, out);
  chamfer_prep<<<(B * N + 255) / 256, 256, 0, stream>>>(pred, ptr_trans, B * N);
  chamfer_prep<<<(B * M + 255) / 256, 256, 0, stream>>>(target, tgt_trans, B * M);

  const int rb1 = N / TILE, rb2 = M / TILE;
  const int waves1 = B * rb1 * CHUNKS;  // 8192
  const int waves2 = B * rb2 * CHUNKS;

  // pass 1: pred rows vs target cols -> dist1 sq-mins
  chamfer_min_pass<<<waves1 / 8, 256, 0, stream>>>(pred, tgt_trans, minbuf, N, M, rb1);
  // pass 2: target rows vs pred cols -> dist2 sq-mins
  chamfer_min_pass<<<waves2 / 8, 256, 0, stream>>>(target, ptr_trans,
                                                   minbuf + (size_t)B * N, M, N, rb2);

  chamfer_finalize<<<(total + 255) / 256, 256, 0, stream>>>(
      minbuf, out, B * N, total, 1.0f / (B * N), 1.0f / (B * M));
}